// SNN_36979668418880
// MI455X (gfx1250) — compile-verified
//
#include <hip/hip_runtime.h>
#include <hip/hip_bf16.h>

typedef __attribute__((ext_vector_type(2))) float v2f;
typedef __attribute__((ext_vector_type(4))) float v4f;
typedef __attribute__((ext_vector_type(8))) float v8f;

#define BATCH 4096
#define NI    250
#define NH    1024
#define NO    5
#define NSTEP 50
#define BETA  0.9f
#define THR   0.5f

// ---------------------------------------------------------------------------
// Kernel A: cur1 = x @ W1^T + b1   (4096x250 * 250x1024 -> 4096x1024)
// One wave per 16x16 tile via V_WMMA_F32_16X16X4_F32.
// K = 250 = 62*4 + 2: main loop (k < 248) is unguarded b64 loads; one peeled
// tail WMMA where only the koff==0 half-wave loads k={248,249}.
// cur1 stores stay RT (default) so Kernel B's reads hit L2 (16.8 MB << 192 MB).
// ---------------------------------------------------------------------------
__global__ __launch_bounds__(128)
void snn_gemm_cur1(const float* __restrict__ x,
                   const float* __restrict__ W1,
                   const float* __restrict__ b1,
                   float* __restrict__ cur1)
{
    const int lane  = threadIdx.x;                   // 0..31
    const int ntile = blockIdx.x * 4 + threadIdx.y;  // 0..63
    const int mtile = blockIdx.y;                    // 0..255
    const int m0 = mtile * 16;
    const int n0 = ntile * 16;

    const int mn   = lane & 15;        // A row / B col for this lane
    const int koff = (lane >> 4) * 2;  // K sub-pair: 0 (lanes 0-15) or 2 (16-31)

    // Row stride = 250 floats = 1000 B (8B aligned); koff adds 0 or 8 B.
    const float* xrow = x  + (size_t)(m0 + mn) * NI + koff;
    const float* wrow = W1 + (size_t)(n0 + mn) * NI + koff;

    v8f c = {};
#pragma unroll 2
    for (int k = 0; k < 248; k += 4) {
        v2f a = *(const v2f*)(xrow + k);
        v2f b = *(const v2f*)(wrow + k);
        c = __builtin_amdgcn_wmma_f32_16x16x4_f32(
                false, a, false, b, (short)0, c, false, false);
    }
    // Tail K-slice: k = 248..251; only k=248,249 exist (koff==0 half-wave).
    {
        v2f a = {0.0f, 0.0f};
        v2f b = {0.0f, 0.0f};
        if (koff == 0) {
            a = *(const v2f*)(xrow + 248);
            b = *(const v2f*)(wrow + 248);
        }
        c = __builtin_amdgcn_wmma_f32_16x16x4_f32(
                false, a, false, b, (short)0, c, false, false);
    }

    // D layout: VGPR i, lanes 0-15 -> M=i, N=lane; lanes 16-31 -> M=8+i, N=lane-16
    const int n  = n0 + (lane & 15);
    const int mb = m0 + (lane >> 4) * 8;
    const float bias = b1[n];
#pragma unroll
    for (int i = 0; i < 8; ++i) {
        cur1[(size_t)(mb + i) * NH + n] = c[i] + bias;
    }
}

// ---------------------------------------------------------------------------
// Kernel B: fused 50-step LIF recurrence (bandwidth-dominant: 839 MB stores).
// Block = 256 threads = 4 batch rows x 64 threads. Each thread owns 16
// CONTIGUOUS hidden neurons -> mem1_rec streamed with NON-TEMPORAL
// global_store_b128 (write-once data, 4.4x L2 capacity: don't pollute L2).
// cur2 = spk1 @ W2^T reduced with wave32 shfl_xor butterflies + a tiny LDS
// combine of the two waves per row (2 barriers/step).
// ---------------------------------------------------------------------------
__global__ __launch_bounds__(256)
void snn_timeloop(const float* __restrict__ cur1,
                  const float* __restrict__ W2,
                  const float* __restrict__ b2,
                  float* __restrict__ out_spk2,   // [50][4096][5]
                  float* __restrict__ out_mem1,   // [50][4096][1024]
                  float* __restrict__ out_mem2)   // [50][4096][5]
{
    const int t      = threadIdx.x;
    const int row    = t >> 6;               // 0..3
    const int lane64 = t & 63;
    const int gr     = blockIdx.x * 4 + row; // global batch row
    const int hb     = lane64 * 16;          // first hidden neuron of thread

    __shared__ float red[8 * NO];            // one slot per wave (8 waves)

    // Preload cur1 slice and W2 slices (vectorized b128 loads).
    float cur[16], m1[16];
    float w2r[NO][16];
    const float* cbase = cur1 + (size_t)gr * NH + hb;
#pragma unroll
    for (int q = 0; q < 4; ++q) {
        v4f v = *(const v4f*)(cbase + 4 * q);
        cur[4*q+0] = v.x; cur[4*q+1] = v.y; cur[4*q+2] = v.z; cur[4*q+3] = v.w;
    }
#pragma unroll
    for (int o = 0; o < NO; ++o) {
        const float* wb = W2 + (size_t)o * NH + hb;
#pragma unroll
        for (int q = 0; q < 4; ++q) {
            v4f v = *(const v4f*)(wb + 4 * q);
            w2r[o][4*q+0] = v.x; w2r[o][4*q+1] = v.y;
            w2r[o][4*q+2] = v.z; w2r[o][4*q+3] = v.w;
        }
    }
#pragma unroll
    for (int j = 0; j < 16; ++j) m1[j] = 0.0f;

    float m2[NO];
#pragma unroll
    for (int o = 0; o < NO; ++o) m2[o] = 0.0f;

    for (int s = 0; s < NSTEP; ++s) {
        float acc[NO] = {0.f, 0.f, 0.f, 0.f, 0.f};
        float* mem1_out = out_mem1 + (size_t)s * BATCH * NH
                                   + (size_t)gr * NH + hb;

#pragma unroll
        for (int q = 0; q < 4; ++q) {
            v4f v;
#pragma unroll
            for (int e = 0; e < 4; ++e) {
                const int j = 4 * q + e;
                // reset = heaviside(mem_prev - thr)*thr; mem = beta*mem + cur - reset
                const float reset = (m1[j] > THR) ? THR : 0.0f;
                m1[j] = BETA * m1[j] + cur[j] - reset;
                // spk1 = heaviside(mem - thr); accumulate spk1 @ W2^T
                if (m1[j] > THR) {
#pragma unroll
                    for (int o = 0; o < NO; ++o) acc[o] += w2r[o][j];
                }
            }
            v.x = m1[4*q+0]; v.y = m1[4*q+1]; v.z = m1[4*q+2]; v.w = m1[4*q+3];
            // Streaming write-once data: non-temporal b128 store (TH=NT).
            __builtin_nontemporal_store(v, (v4f*)(mem1_out + 4 * q));
        }

        // Intra-wave (32-lane) butterfly reduction, register-only.
#pragma unroll
        for (int off = 16; off > 0; off >>= 1) {
#pragma unroll
            for (int o = 0; o < NO; ++o)
                acc[o] += __shfl_xor(acc[o], off, 32);
        }
        // Two waves per row -> combine through LDS.
        if ((t & 31) == 0) {
#pragma unroll
            for (int o = 0; o < NO; ++o) red[(t >> 5) * NO + o] = acc[o];
        }
        __syncthreads();

        if (lane64 == 0) {
            float* spk2_out = out_spk2 + (size_t)s * BATCH * NO + (size_t)gr * NO;
            float* mem2_out = out_mem2 + (size_t)s * BATCH * NO + (size_t)gr * NO;
#pragma unroll
            for (int o = 0; o < NO; ++o) {
                const float cur2 = red[(row * 2) * NO + o]
                                 + red[(row * 2 + 1) * NO + o] + b2[o];
                const float reset2 = (m2[o] > THR) ? THR : 0.0f;
                m2[o] = BETA * m2[o] + cur2 - reset2;
                const float spk2 = (m2[o] > THR) ? 1.0f : 0.0f;
                __builtin_nontemporal_store(spk2, spk2_out + o);
                __builtin_nontemporal_store(m2[o], mem2_out + o);
            }
        }
        __syncthreads();   // red reused next step
    }
}

extern "C" void kernel_launch(void* const* d_in, const int* in_sizes, int n_in,
                              void* d_out, int out_size, void* d_ws, size_t ws_size,
                              hipStream_t stream)
{
    const float* x  = (const float*)d_in[0];   // [4096][250]
    const float* W1 = (const float*)d_in[1];   // [1024][250]
    const float* b1 = (const float*)d_in[2];   // [1024]
    const float* W2 = (const float*)d_in[3];   // [5][1024]
    const float* b2 = (const float*)d_in[4];   // [5]

    float* out = (float*)d_out;
    // Return order: spk2_rec, mem1_rec, mem2_rec (flattened, concatenated)
    float* out_spk2 = out;                                       // 50*4096*5
    float* out_mem1 = out + (size_t)NSTEP * BATCH * NO;          // 50*4096*1024
    float* out_mem2 = out_mem1 + (size_t)NSTEP * BATCH * NH;     // 50*4096*5

    float* cur1 = (float*)d_ws;                                  // 4096*1024 f32

    // Kernel A: cur1 = x @ W1^T + b1
    dim3 gridA(NH / 64, BATCH / 16);   // (16, 256): 4 N-tiles per block
    dim3 blockA(32, 4);
    snn_gemm_cur1<<<gridA, blockA, 0, stream>>>(x, W1, b1, cur1);

    // Kernel B: fused 50-step recurrence
    snn_timeloop<<<BATCH / 4, 256, 0, stream>>>(cur1, W2, b2,
                                                out_spk2, out_mem1, out_mem2);
}